// ZCANormSVDPI_Cluster_37151467110666
// MI455X (gfx1250) — compile-verified
//
#include <hip/hip_runtime.h>
#include <hip/hip_bf16.h>
#include <math.h>

// ZCANormSVDPI: batched ZCA whitening (B=4096, C=64, M=128, fp32)
// Heavy GEMMs (cov Gram + apply) use V_WMMA_F32_16X16X4_F32 (CDNA5 fp32 matrix op).

typedef float v2f __attribute__((ext_vector_type(2)));
typedef float v8f __attribute__((ext_vector_type(8)));

#define BDIM 4096
#define CDIM 64
#define MDIM 128
#define ZEPS 1.0e-4f

// ---------------------------------------------------------------------------
// 1) mu[c][m] = mean over b of x[b][c][m]           (coalesced over c*M+m)
// ---------------------------------------------------------------------------
__global__ __launch_bounds__(256) void zca_mean(const float* __restrict__ x,
                                                float* __restrict__ mu) {
  int idx = blockIdx.x * 256 + threadIdx.x;           // 0..8191 = c*128+m
  float s = 0.0f;
  for (int b = 0; b < BDIM; ++b)
    s += x[(size_t)b * (CDIM * MDIM) + idx];
  mu[idx] = s * (1.0f / BDIM);
}

// ---------------------------------------------------------------------------
// 2) cov[m] = (Y Y^T)/B - mu mu^T + eps*I,  Y[c][b] = x[b][c][m]
//    one block per m; 16 waves; each wave owns one 16x16 tile of the 64x64
//    output; K=4096 swept in 64-wide LDS chunks via wmma_f32_16x16x4.
// ---------------------------------------------------------------------------
__global__ __launch_bounds__(512) void zca_gram_cov(const float* __restrict__ x,
                                                    const float* __restrict__ mu,
                                                    float* __restrict__ covG) {
  const int m    = blockIdx.x;
  const int tid  = threadIdx.x;
  const int wave = tid >> 5;
  const int lane = tid & 31;
  const int lh   = lane & 15;
  const bool hi  = lane >= 16;
  const int ti   = wave >> 2;   // row tile 0..3
  const int tj   = wave & 3;    // col tile 0..3

  __shared__ float Ys[64][65];  // padded: stride 65 avoids bank conflicts

  v8f acc = {};
  for (int bb = 0; bb < BDIM; bb += 64) {
    __syncthreads();
    for (int it = 0; it < 8; ++it) {
      int li = tid + it * 512;
      int c  = li & 63;
      int bl = li >> 6;
      Ys[c][bl] = x[(size_t)(bb + bl) * (CDIM * MDIM) + (size_t)c * MDIM + m];
    }
    if (bb + 64 < BDIM) {   // global_prefetch_b8 of next chunk
      int c = tid & 63, bl = tid >> 6;
      __builtin_prefetch(&x[(size_t)(bb + 64 + bl) * (CDIM * MDIM) + (size_t)c * MDIM + m], 0, 1);
    }
    __syncthreads();
#pragma unroll
    for (int k = 0; k < 64; k += 4) {
      int kk = k + (hi ? 2 : 0);
      v2f a, b;
      // A (16x4): rows 16*ti.., K striped lanes0-15:K0/K1, lanes16-31:K2/K3
      a.x = Ys[ti * 16 + lh][kk];
      a.y = Ys[ti * 16 + lh][kk + 1];
      // B (4x16) = Y^T tile: B[K][N] = Y[16*tj+N][k+K]
      b.x = Ys[tj * 16 + lh][kk];
      b.y = Ys[tj * 16 + lh][kk + 1];
      acc = __builtin_amdgcn_wmma_f32_16x16x4_f32(false, a, false, b,
                                                  (short)0, acc, false, false);
    }
  }
  const float invB = 1.0f / BDIM;
#pragma unroll
  for (int vv = 0; vv < 8; ++vv) {
    int row = ti * 16 + vv + (hi ? 8 : 0);
    int col = tj * 16 + lh;
    float mur = mu[row * MDIM + m];
    float muc = mu[col * MDIM + m];
    float val = acc[vv] * invB - mur * muc + (row == col ? ZEPS : 0.0f);
    covG[(size_t)m * 4096 + row * 64 + col] = val;
  }
}

// ---------------------------------------------------------------------------
// 3) shrinkage + cyclic Jacobi eigensolver (round-robin, 32 disjoint pairs
//    per round). Writes shrunk cov back, eigvals L (desc) and eigvecs U
//    (columns, desc), and zeroes S_hat.
// ---------------------------------------------------------------------------
__global__ __launch_bounds__(256) void zca_shrink_jacobi(float* __restrict__ covG,
                                                         float* __restrict__ Ug,
                                                         float* __restrict__ Lg,
                                                         float* __restrict__ SG) {
  const int m = blockIdx.x;
  const int tid = threadIdx.x;
  __shared__ float A[64][65];
  __shared__ float Vv[64][65];
  __shared__ float red1[256], red2[256];
  __shared__ float cs[32], sn[32];
  __shared__ int   pidx[32], qidx[32];
  __shared__ int   arr[64];
  __shared__ int   rankof[64];
  __shared__ float s_rho, s_tr;

  float sq = 0.0f, dg = 0.0f;
  for (int it = 0; it < 16; ++it) {
    int li = tid + it * 256;
    int r = li >> 6, c = li & 63;
    float v = covG[(size_t)m * 4096 + li];
    A[r][c]  = v;
    Vv[r][c] = (r == c) ? 1.0f : 0.0f;
    SG[(size_t)m * 4096 + li] = 0.0f;      // init S_hat = 0
    sq += v * v;
    if (r == c) dg += v;
  }
  red1[tid] = sq; red2[tid] = dg;
  __syncthreads();
  for (int s = 128; s > 0; s >>= 1) {
    if (tid < s) { red1[tid] += red1[tid + s]; red2[tid] += red2[tid + s]; }
    __syncthreads();
  }
  if (tid == 0) {
    float tr  = red2[0] / 64.0f;                       // mean diag
    float st  = red1[0] / 64.0f;                       // sum(cov^2)/C
    float num = (64.0f * tr) * (64.0f * tr) - st;
    float den = (BDIM - 1.0f) * (st - tr * tr);
    s_rho = fminf(num / den, 1.0f);
    s_tr  = tr;
  }
  __syncthreads();
  float rho = s_rho, tr = s_tr;
  for (int it = 0; it < 16; ++it) {
    int li = tid + it * 256;
    int r = li >> 6, c = li & 63;
    float v = (1.0f - rho) * A[r][c] + ((r == c) ? rho * tr : 0.0f);
    A[r][c] = v;
    covG[(size_t)m * 4096 + li] = v;       // shrunk cov persisted for scan
  }
  if (tid < 64) arr[tid] = tid;
  __syncthreads();

  for (int sweep = 0; sweep < 10; ++sweep) {
    for (int round = 0; round < 63; ++round) {
      if (tid < 32) {
        int p = arr[tid], q = arr[63 - tid];
        if (p > q) { int t2 = p; p = q; q = t2; }
        pidx[tid] = p; qidx[tid] = q;
        float apq = A[p][q];
        float c = 1.0f, s = 0.0f;
        if (fabsf(apq) > 1e-12f) {
          float app = A[p][p], aqq = A[q][q];
          float tau = (aqq - app) / (2.0f * apq);
          float t = ((tau >= 0.0f) ? 1.0f : -1.0f) /
                    (fabsf(tau) + sqrtf(1.0f + tau * tau));
          c = rsqrtf(1.0f + t * t);
          s = t * c;
        }
        cs[tid] = c; sn[tid] = s;
      }
      __syncthreads();
      // column updates on A and V (pairs are disjoint -> race-free)
      for (int it = 0; it < 8; ++it) {
        int item = tid + it * 256;            // 0..2047
        int pr = item >> 6, row = item & 63;
        int p = pidx[pr], q = qidx[pr];
        float c = cs[pr], s = sn[pr];
        float aip = A[row][p], aiq = A[row][q];
        A[row][p] = c * aip - s * aiq;
        A[row][q] = s * aip + c * aiq;
        float vip = Vv[row][p], viq = Vv[row][q];
        Vv[row][p] = c * vip - s * viq;
        Vv[row][q] = s * vip + c * viq;
      }
      __syncthreads();
      // row updates on A
      for (int it = 0; it < 8; ++it) {
        int item = tid + it * 256;
        int pr = item >> 6, col = item & 63;
        int p = pidx[pr], q = qidx[pr];
        float c = cs[pr], s = sn[pr];
        float apj = A[p][col], aqj = A[q][col];
        A[p][col] = c * apj - s * aqj;
        A[q][col] = s * apj + c * aqj;
      }
      __syncthreads();
      if (tid == 0) {                        // rotate tournament schedule
        int tmp = arr[63];
        for (int k2 = 63; k2 >= 2; --k2) arr[k2] = arr[k2 - 1];
        arr[1] = tmp;
      }
      __syncthreads();
    }
  }

  // sort eigenpairs descending (rank by selection)
  if (tid < 64) {
    float le = A[tid][tid];
    int cnt = 0;
    for (int u = 0; u < 64; ++u) {
      float lu = A[u][u];
      cnt += (lu > le) || (lu == le && u < tid);
    }
    rankof[tid] = cnt;
    Lg[m * 64 + cnt] = le;
  }
  __syncthreads();
  for (int it = 0; it < 16; ++it) {
    int li = tid + it * 256;
    int row = li >> 6, src = li & 63;
    Ug[(size_t)m * 4096 + row * 64 + rankof[src]] = Vv[row][src];
  }
}

// ---------------------------------------------------------------------------
// 4) counter: global over all 128 m (reference: any over batch axis)
//    flags[0]=counter, flags[1]=active, flags[2]=ok_current
// ---------------------------------------------------------------------------
__global__ void zca_counter(const float* __restrict__ Lg, int* __restrict__ flags) {
  __shared__ unsigned badm[2];
  if (threadIdx.x < 2) badm[threadIdx.x] = 0u;
  __syncthreads();
  int m = threadIdx.x;
  if (m < 128) {
    float tot = 0.0f;
    for (int j = 0; j < 64; ++j) tot += Lg[m * 64 + j];
    float run = 0.0f;
    for (int j = 0; j < 64; ++j) {
      float l = Lg[m * 64 + j];
      run += l;
      if ((run / tot >= 1.0f - ZEPS) || (l <= ZEPS))
        atomicOr(&badm[j >> 5], 1u << (j & 31));
    }
  }
  __syncthreads();
  if (threadIdx.x == 0) {
    int counter = 64;
    for (int j = 0; j < 64; ++j)
      if ((badm[j >> 5] >> (j & 31)) & 1u) { counter = j; break; }
    flags[0] = counter;
    flags[1] = 1;
    flags[2] = 0;
  }
}

// ---------------------------------------------------------------------------
// 5a) per-m Rayleigh quotient for step j: w = cov_j v,  lam = v.w / v.v
//     (v = row j of U, matching the reference's indexing)
// ---------------------------------------------------------------------------
__global__ __launch_bounds__(64) void zca_lam(const float* __restrict__ covG,
                                              const float* __restrict__ Ug,
                                              float* __restrict__ lamG,
                                              float* __restrict__ wG, int j) {
  const int m = blockIdx.x, t = threadIdx.x;
  __shared__ float vsh[64], wsh[64];
  vsh[t] = Ug[(size_t)m * 4096 + j * 64 + t];
  __syncthreads();
  float acc = 0.0f;
  const float* row = covG + (size_t)m * 4096 + t * 64;
  for (int c2 = 0; c2 < 64; ++c2) acc += row[c2] * vsh[c2];
  wsh[t] = acc;
  wG[m * 64 + t] = acc;
  __syncthreads();
  if (t == 0) {
    float num = 0.0f, den = 0.0f;
    for (int c2 = 0; c2 < 64; ++c2) { num += vsh[c2] * wsh[c2]; den += vsh[c2] * vsh[c2]; }
    lamG[m] = num / den;
  }
}

// 5b) global ok decision for step j (any() across all m, chained active flag)
__global__ void zca_ok(const float* __restrict__ lamG, const float* __restrict__ Lg,
                       int* __restrict__ flags, int j) {
  __shared__ int bad;
  if (threadIdx.x == 0) bad = 0;
  __syncthreads();
  int m = threadIdx.x;
  if (m < 128) {
    float lam = lamG[m];
    float Lj  = Lg[m * 64 + j];
    if (lam < 0.0f || fabsf(lam - Lj) / Lj > 0.1f) atomicOr(&bad, 1);
  }
  __syncthreads();
  if (threadIdx.x == 0) {
    int ok = (flags[1] != 0) && (j < flags[0]) && (bad == 0);
    flags[1] = ok;
    flags[2] = ok;
  }
}

// 5c) conditional rank-1 updates: S += rsqrt(lam) v v^T ; cov -= w v^T
__global__ __launch_bounds__(256) void zca_deflate(float* __restrict__ covG,
                                                   float* __restrict__ SG,
                                                   const float* __restrict__ Ug,
                                                   const float* __restrict__ lamG,
                                                   const float* __restrict__ wG,
                                                   const int* __restrict__ flags, int j) {
  if (flags[2] == 0) return;                  // uniform branch
  const int m = blockIdx.x, tid = threadIdx.x;
  __shared__ float vsh[64], wsh[64];
  __shared__ float srs;
  if (tid < 64) {
    vsh[tid] = Ug[(size_t)m * 4096 + j * 64 + tid];
    wsh[tid] = wG[m * 64 + tid];
  }
  if (tid == 0) srs = rsqrtf(lamG[m]);
  __syncthreads();
  for (int it = 0; it < 16; ++it) {
    int li = tid + it * 256;
    int r = li >> 6, c = li & 63;
    size_t idx = (size_t)m * 4096 + li;
    SG[idx]   += srs * vsh[r] * vsh[c];
    covG[idx] -= wsh[r] * vsh[c];
  }
}

// ---------------------------------------------------------------------------
// 6) out[b][c][m] = sum_k S[m][c][k] * (x[b][k][m] - mu[k][m])   via WMMA
// ---------------------------------------------------------------------------
__global__ __launch_bounds__(512) void zca_apply(const float* __restrict__ x,
                                                 const float* __restrict__ mu,
                                                 const float* __restrict__ SG,
                                                 float* __restrict__ out) {
  const int m    = blockIdx.x;
  const int tid  = threadIdx.x;
  const int wave = tid >> 5;
  const int lane = tid & 31;
  const int lh   = lane & 15;
  const bool hi  = lane >= 16;
  const int ti   = wave >> 2;
  const int tj   = wave & 3;

  __shared__ float Ss[64][65];
  __shared__ float Yl[64][65];
  __shared__ float mus[64];

  for (int it = 0; it < 8; ++it) {
    int li = tid + it * 512;
    Ss[li >> 6][li & 63] = SG[(size_t)m * 4096 + li];
  }
  if (tid < 64) mus[tid] = mu[tid * MDIM + m];
  __syncthreads();

  for (int bb = 0; bb < BDIM; bb += 64) {
    for (int it = 0; it < 8; ++it) {
      int li = tid + it * 512;
      int c = li & 63, bl = li >> 6;
      Yl[c][bl] = x[(size_t)(bb + bl) * (CDIM * MDIM) + (size_t)c * MDIM + m] - mus[c];
    }
    if (bb + 64 < BDIM) {
      int c = tid & 63, bl = tid >> 6;
      __builtin_prefetch(&x[(size_t)(bb + 64 + bl) * (CDIM * MDIM) + (size_t)c * MDIM + m], 0, 1);
    }
    __syncthreads();
    v8f acc = {};
#pragma unroll
    for (int k = 0; k < 64; k += 4) {
      int kk = k + (hi ? 2 : 0);
      v2f a, b;
      a.x = Ss[ti * 16 + lh][kk];         // A = S rows
      a.y = Ss[ti * 16 + lh][kk + 1];
      b.x = Yl[kk][tj * 16 + lh];         // B[K][N] = Yc[k+K][16*tj+N]
      b.y = Yl[kk + 1][tj * 16 + lh];
      acc = __builtin_amdgcn_wmma_f32_16x16x4_f32(false, a, false, b,
                                                  (short)0, acc, false, false);
    }
#pragma unroll
    for (int vv = 0; vv < 8; ++vv) {
      int row  = ti * 16 + vv + (hi ? 8 : 0);
      int bcol = bb + tj * 16 + lh;
      out[(size_t)bcol * (CDIM * MDIM) + (size_t)row * MDIM + m] = acc[vv];
    }
    __syncthreads();
  }
}

// ---------------------------------------------------------------------------
extern "C" void kernel_launch(void* const* d_in, const int* in_sizes, int n_in,
                              void* d_out, int out_size, void* d_ws, size_t ws_size,
                              hipStream_t stream) {
  (void)in_sizes; (void)n_in; (void)out_size; (void)ws_size;
  const float* x = (const float*)d_in[0];
  float* out = (float*)d_out;

  float* ws   = (float*)d_ws;
  float* mu   = ws;                 //   8192 floats
  float* covG = mu + 8192;          // 524288
  float* Ug   = covG + 524288;      // 524288
  float* Lg   = Ug + 524288;        //   8192
  float* SG   = Lg + 8192;          // 524288
  float* lamG = SG + 524288;        //    128
  float* wG   = lamG + 128;         //   8192
  int* flags  = (int*)(wG + 8192);  // counter/active/okcur  (~6.4 MB total)

  zca_mean<<<(CDIM * MDIM) / 256, 256, 0, stream>>>(x, mu);
  zca_gram_cov<<<MDIM, 512, 0, stream>>>(x, mu, covG);
  zca_shrink_jacobi<<<MDIM, 256, 0, stream>>>(covG, Ug, Lg, SG);
  zca_counter<<<1, 128, 0, stream>>>(Lg, flags);
  for (int j = 0; j < 64; ++j) {
    zca_lam<<<MDIM, 64, 0, stream>>>(covG, Ug, lamG, wG, j);
    zca_ok<<<1, 128, 0, stream>>>(lamG, Lg, flags, j);
    zca_deflate<<<MDIM, 256, 0, stream>>>(covG, SG, Ug, lamG, wG, flags, j);
  }
  zca_apply<<<MDIM, 512, 0, stream>>>(x, mu, SG, out);
}